// PonitaFixedSize_38663295598622
// MI455X (gfx1250) — compile-verified
//
#include <hip/hip_runtime.h>
#include <math.h>
#include <stdint.h>

// ---------------------------------------------------------------------------
// Ponita fixed-size forward, CDNA5 (gfx1250) wave32 + WMMA f32 16x16x4
//   + async global->LDS double-buffered staging (ASYNCcnt)
// B=1, N=256, O=16 orientations, H=64, BD=32, hidden=64, WF*H=256, L=2
// ---------------------------------------------------------------------------

#define NPT   256   // points
#define NORI  16    // orientations
#define HD    64    // feature width
#define BDIM  32    // basis dim
#define HIDB  64    // basis MLP hidden
#define WFH   256   // lin1 width
#define NLAY  2
#define NCH   (NPT / 16)   // n chunks

typedef __attribute__((ext_vector_type(2))) float v2f;
typedef __attribute__((ext_vector_type(8))) float v8f;

__device__ __forceinline__ v8f splat8(float x) {
  v8f r;
#pragma unroll
  for (int i = 0; i < 8; ++i) r[i] = x;
  return r;
}

// D = A(16x4) * B(4x16) + C  -- fp32 WMMA, wave32
__device__ __forceinline__ v8f wmma4(v2f a, v2f b, v8f c) {
  return __builtin_amdgcn_wmma_f32_16x16x4_f32(
      /*neg_a=*/false, a, /*neg_b=*/false, b,
      /*c_mod=*/(short)0, c, /*reuse_a=*/false, /*reuse_b=*/false);
}

__device__ __forceinline__ float gelu_tanh(float x) {
  // jax.nn.gelu(approximate=True)
  float x3 = x * x * x;
  return 0.5f * x * (1.0f + tanhf(0.7978845608028654f * (x + 0.044715f * x3)));
}

// async copy of 16 bytes: from the global address gaddr to the LDS address lds_off   (per active lane)
__device__ __forceinline__ void async_b128(uint32_t lds_off, const float* gaddr) {
  uint64_t ga = (uint64_t)(uintptr_t)gaddr;
  asm volatile("global_load_async_to_lds_b128 %0, %1, off"
               :: "v"(lds_off), "v"(ga) : "memory");
}

// ---------------------------------------------------------------------------
// K1: fiber basis (O*O rows, MLP 3->64->32) + rk[l][p][o][c] projection
// ---------------------------------------------------------------------------
__global__ __launch_bounds__(256) void fiber_kernel(
    const float* __restrict__ rkbW1, const float* __restrict__ rkbB1,
    const float* __restrict__ rkbW2, const float* __restrict__ rkbB2,
    const float* __restrict__ convRW,   // (L, 32, 64)
    float* __restrict__ rk)             // (L, 16, 16, 64)
{
  __shared__ float fbs[NORI * NORI * BDIM];  // 32 KB
  const int t = threadIdx.x;
  const int p = t >> 4, o = t & 15;
  const float thp = 6.283185307179586f * (float)p / 16.0f;
  const float tho = 6.283185307179586f * (float)o / 16.0f;
  const float z  = cosf(thp) * cosf(tho) + sinf(thp) * sinf(tho);
  const float f1 = z, f2 = z * z, f3 = f2 * z;

  float hid[HIDB];
#pragma unroll 4
  for (int j = 0; j < HIDB; ++j)
    hid[j] = gelu_tanh(f1 * rkbW1[j] + f2 * rkbW1[64 + j] + f3 * rkbW1[128 + j] + rkbB1[j]);

  for (int d = 0; d < BDIM; ++d) {
    float a = rkbB2[d];
    for (int j = 0; j < HIDB; ++j) a += hid[j] * rkbW2[j * BDIM + d];
    fbs[(p * NORI + o) * BDIM + d] = gelu_tanh(a);
  }
  __syncthreads();

  for (int idx = t; idx < NLAY * NORI * NORI * HD; idx += 256) {
    const int c = idx & 63;
    int rest = idx >> 6;
    const int oo = rest & 15; rest >>= 4;
    const int pp = rest & 15;
    const int l  = rest >> 4;
    const float* fbr = &fbs[(pp * NORI + oo) * BDIM];
    const float* w   = convRW + l * (BDIM * HD);
    float a = 0.0f;
#pragma unroll 8
    for (int d = 0; d < BDIM; ++d) a += fbr[d] * w[d * HD + c];
    rk[idx] = a;
  }
}

// ---------------------------------------------------------------------------
// K2: embedding  h[n,o,c] = (x @ emb_W)[n,c]  broadcast over o
// ---------------------------------------------------------------------------
__global__ __launch_bounds__(256) void embed_kernel(
    const float* __restrict__ x, const float* __restrict__ embW,
    float* __restrict__ h)
{
  const int idx = blockIdx.x * 256 + threadIdx.x;   // < N*O*H = 262144
  const int c = idx & 63;
  const int n = idx >> 10;                           // / (O*H)
  float a = x[n * 3 + 0] * embW[0 * HD + c]
          + x[n * 3 + 1] * embW[1 * HD + c]
          + x[n * 3 + 2] * embW[2 * HD + c];
  h[idx] = a;
}

// ---------------------------------------------------------------------------
// K3 (per layer): fused spatial conv.
//   one wave per (m, o):  c[m,o,:] = sum_n (MLP(poly(inv(m,n,o))) @ sW) .* h[n,o,:]
// Block = 4 waves sharing o (consecutive m). h[n-chunk, o, :] staged into LDS
// by async DMA (double-buffered); WMMA chain per 16-row n chunk:
//   P(16x16) @ W1(16x64) -> gelu -> @ W2(64x32) -> gelu -> @ sW(32x64)
// ---------------------------------------------------------------------------
__global__ __launch_bounds__(128) void conv_kernel(
    const float* __restrict__ pos,
    const float* __restrict__ skbW1, const float* __restrict__ skbB1,
    const float* __restrict__ skbW2, const float* __restrict__ skbB2,
    const float* __restrict__ convSW,       // (32,64), layer slice
    const float* __restrict__ hbuf,         // (N,O,H)
    float* __restrict__ cbuf)               // (N,O,H)
{
  __shared__ float sW1[16][HD];             // padded 14->16 rows
  __shared__ float sB1[HD];
  __shared__ float sW2[HIDB][BDIM];
  __shared__ float sB2[BDIM];
  __shared__ float sSW[BDIM][HD];
  __shared__ float sPos[NPT][2];
  __shared__ float zscr[4][16][68];         // per-wave Z1 (layout transpose)
  __shared__ float kscr[4][16][36];         // per-wave kb
  __shared__ __align__(16) float hstage[2][16][68];  // async-staged h chunks

  const int tid = threadIdx.x;
  const int lane = tid & 31, wave = tid >> 5;
  const int row  = lane & 15, half = lane >> 4;
  const int o    = blockIdx.x & 15;
  const int m    = (blockIdx.x >> 4) * 4 + wave;

  // ---- stage 0 kicked off by wave 0 immediately (overlaps weight loads)
  const int cl4 = (lane & 15) * 4;   // column group for async lanes
  if (wave == 0) {
#pragma unroll
    for (int j = 0; j < 8; ++j) {
      const int r = 2 * j + half;    // chunk-local row
      const uint32_t la = (uint32_t)(uintptr_t)&hstage[0][r][cl4];
      async_b128(la, hbuf + ((r * NORI + o) * HD + cl4));
    }
  }

  for (int i = tid; i < 16 * HD; i += 128) {
    const int r = i >> 6, c = i & 63;
    sW1[r][c] = (r < 14) ? skbW1[r * HD + c] : 0.0f;
  }
  for (int i = tid; i < HD; i += 128) sB1[i] = skbB1[i];
  for (int i = tid; i < HIDB * BDIM; i += 128) sW2[i >> 5][i & 31] = skbW2[i];
  if (tid < BDIM) sB2[tid] = skbB2[tid];
  for (int i = tid; i < BDIM * HD; i += 128) sSW[i >> 6][i & 63] = convSW[i];
  for (int i = tid; i < NPT * 2; i += 128) sPos[i >> 1][i & 1] = pos[i];
  __syncthreads();

  const float th = 6.283185307179586f * (float)o / 16.0f;
  const float ox = cosf(th), oy = sinf(th);
  const float pmx = sPos[m][0], pmy = sPos[m][1];

  float* z1 = &zscr[wave][0][0];   // [16][68]
  float* kb = &kscr[wave][0][0];   // [16][36]

  v8f csum[4];
#pragma unroll
  for (int t = 0; t < 4; ++t) csum[t] = splat8(0.0f);

  for (int nb = 0; nb < NCH; ++nb) {
    const int n0  = nb * 16;
    const int cur = nb & 1;

    // ---- kick off async DMA for the NEXT chunk into the other buffer
    if (nb + 1 < NCH && wave == 0) {
#pragma unroll
      for (int j = 0; j < 8; ++j) {
        const int r = 2 * j + half;
        const uint32_t la = (uint32_t)(uintptr_t)&hstage[1 - cur][r][cl4];
        async_b128(la, hbuf + (((n0 + 16 + r) * NORI + o) * HD + cl4));
      }
    }
    // ---- current chunk ready: <=8 outstanding means chunk nb has landed
    if (wave == 0) {
      if (nb + 1 < NCH) asm volatile("s_wait_asynccnt 0x8" ::: "memory");
      else              asm volatile("s_wait_asynccnt 0x0" ::: "memory");
    }
    __syncthreads();

    // --- per-lane spatial invariants + poly features (row n0+row, dup halves)
    const float rx = sPos[n0 + row][0] - pmx;
    const float ry = sPos[n0 + row][1] - pmy;
    const float i1 = rx * ox + ry * oy;                  // inv1
    const float rn = sqrtf(rx * rx + ry * ry);
    const float i2 = rn * fabsf(1.0f - i1);              // inv2 (per reference)
    float f[16];
    f[0] = i1;        f[1] = i2;
    f[2] = i1 * i1;   f[3] = i1 * i2;  f[4] = i2 * i1;  f[5] = i2 * i2;
    f[6] = f[2] * i1; f[7] = f[2] * i2;
    f[8] = f[3] * i1; f[9] = f[3] * i2;
    f[10] = f[4] * i1; f[11] = f[4] * i2;
    f[12] = f[5] * i1; f[13] = f[5] * i2;
    f[14] = 0.0f;     f[15] = 0.0f;

    // --- GEMM1: Z1 = gelu(P @ W1 + b1), 4 col-tiles x 4 k-steps
#pragma unroll
    for (int t = 0; t < 4; ++t) {
      const int cb = t * 16;
      v8f acc = splat8(sB1[cb + row]);
#pragma unroll
      for (int kk = 0; kk < 4; ++kk) {
        v2f a; a.x = f[4 * kk + 2 * half];            a.y = f[4 * kk + 2 * half + 1];
        v2f b; b.x = sW1[4 * kk + 2 * half][cb + row]; b.y = sW1[4 * kk + 2 * half + 1][cb + row];
        acc = wmma4(a, b, acc);
      }
#pragma unroll
      for (int v = 0; v < 8; ++v)
        z1[(v + 8 * half) * 68 + cb + row] = gelu_tanh(acc[v]);
    }

    // --- GEMM2: kb = gelu(Z1 @ W2 + b2), 2 col-tiles x 16 k-steps
#pragma unroll
    for (int t = 0; t < 2; ++t) {
      const int cb = t * 16;
      v8f acc = splat8(sB2[cb + row]);
#pragma unroll
      for (int kk = 0; kk < 16; ++kk) {
        v2f a; a.x = z1[row * 68 + 4 * kk + 2 * half]; a.y = z1[row * 68 + 4 * kk + 2 * half + 1];
        v2f b; b.x = sW2[4 * kk + 2 * half][cb + row]; b.y = sW2[4 * kk + 2 * half + 1][cb + row];
        acc = wmma4(a, b, acc);
      }
#pragma unroll
      for (int v = 0; v < 8; ++v)
        kb[(v + 8 * half) * 36 + cb + row] = gelu_tanh(acc[v]);
    }

    // --- GEMM3: sk = kb @ sW, then c[:,c] += sk[n,c] * h[n,o,c] (h from LDS)
#pragma unroll
    for (int t = 0; t < 4; ++t) {
      const int cb = t * 16;
      v8f acc = splat8(0.0f);
#pragma unroll
      for (int kk = 0; kk < 8; ++kk) {
        v2f a; a.x = kb[row * 36 + 4 * kk + 2 * half]; a.y = kb[row * 36 + 4 * kk + 2 * half + 1];
        v2f b; b.x = sSW[4 * kk + 2 * half][cb + row]; b.y = sSW[4 * kk + 2 * half + 1][cb + row];
        acc = wmma4(a, b, acc);
      }
#pragma unroll
      for (int v = 0; v < 8; ++v) {
        const float hv = hstage[cur][v + 8 * half][cb + row];
        csum[t][v] += acc[v] * hv;
      }
    }

    __syncthreads();   // all waves done reading hstage[cur] before it is re-DMA'd
  }

  // reduce over D rows + lane halves, write c[m,o,c]
#pragma unroll
  for (int t = 0; t < 4; ++t) {
    float s = 0.0f;
#pragma unroll
    for (int v = 0; v < 8; ++v) s += csum[t][v];
    s += __shfl_xor(s, 16, 32);
    if (half == 0)
      cbuf[(m * NORI + o) * HD + t * 16 + row] = s;
  }
}

// ---------------------------------------------------------------------------
// K4 (per layer): orientation mix + bias + LayerNorm + MLP(64->256->64) + res
//   one wave per m (16 rows p=0..15)
// ---------------------------------------------------------------------------
__global__ __launch_bounds__(32) void mix_kernel(
    const float* __restrict__ cbuf,   // (N,O,H)
    const float* __restrict__ rk,     // (16,16,64) layer slice
    const float* __restrict__ convB,  // (64)
    const float* __restrict__ lnS, const float* __restrict__ lnB,
    const float* __restrict__ W1, const float* __restrict__ B1,  // (64,256),(256)
    const float* __restrict__ W2, const float* __restrict__ B2,  // (256,64),(64)
    float* __restrict__ hbuf)         // (N,O,H) updated in place
{
  __shared__ float cm[NORI][HD];        // c[m,o,c]
  __shared__ float a2[16][68];          // LN output (MLP input)
  __shared__ float zs[16][260];         // gelu(lin1) scratch

  const int lane = threadIdx.x;
  const int row  = lane & 15, half = lane >> 4;
  const int m    = blockIdx.x;

  for (int i = lane; i < NORI * HD; i += 32) cm[i >> 6][i & 63] = cbuf[m * NORI * HD + i];
  __syncthreads();

  // orientation mixing: c2[p][c] = sum_o cm[o][c] * rk[p][o][c]; then +bias, LN
  const int p = row;
  float vals[32];
  float msum = 0.0f, msq = 0.0f;
  for (int j = 0; j < 32; ++j) {
    const int c = 32 * half + j;
    float acc = convB[c];
#pragma unroll 4
    for (int oo = 0; oo < NORI; ++oo) acc += cm[oo][c] * rk[(p * NORI + oo) * HD + c];
    vals[j] = acc; msum += acc; msq += acc * acc;
  }
  msum += __shfl_xor(msum, 16, 32);
  msq  += __shfl_xor(msq, 16, 32);
  const float mu  = msum * (1.0f / 64.0f);
  const float var = msq * (1.0f / 64.0f) - mu * mu;
  const float inv = rsqrtf(var + 1e-6f);
  for (int j = 0; j < 32; ++j) {
    const int c = 32 * half + j;
    a2[p][c] = (vals[j] - mu) * inv * lnS[c] + lnB[c];
  }
  __syncthreads();

  // lin1: (16x64)@(64x256) -> gelu, 16 col-tiles x 16 k-steps
  for (int t = 0; t < 16; ++t) {
    const int cb = t * 16;
    v8f acc = splat8(B1[cb + row]);
#pragma unroll
    for (int kk = 0; kk < 16; ++kk) {
      v2f a; a.x = a2[row][4 * kk + 2 * half];            a.y = a2[row][4 * kk + 2 * half + 1];
      v2f b; b.x = W1[(4 * kk + 2 * half) * WFH + cb + row];
             b.y = W1[(4 * kk + 2 * half + 1) * WFH + cb + row];
      acc = wmma4(a, b, acc);
    }
#pragma unroll
    for (int v = 0; v < 8; ++v) zs[v + 8 * half][cb + row] = gelu_tanh(acc[v]);
  }
  __syncthreads();

  // lin2: (16x256)@(256x64) + b2, residual into h
  for (int t = 0; t < 4; ++t) {
    const int cb = t * 16;
    v8f acc = splat8(B2[cb + row]);
#pragma unroll 8
    for (int kk = 0; kk < 64; ++kk) {
      v2f a; a.x = zs[row][4 * kk + 2 * half];            a.y = zs[row][4 * kk + 2 * half + 1];
      v2f b; b.x = W2[(4 * kk + 2 * half) * HD + cb + row];
             b.y = W2[(4 * kk + 2 * half + 1) * HD + cb + row];
      acc = wmma4(a, b, acc);
    }
#pragma unroll
    for (int v = 0; v < 8; ++v) {
      const int pp  = v + 8 * half;
      const int idx = (m * NORI + pp) * HD + cb + row;
      hbuf[idx] = hbuf[idx] + acc[v];
    }
  }
}

// ---------------------------------------------------------------------------
// K5: readout  out[k] = mean_{n,o}( h[n,o,:] @ readout_W[:,k] )
// ---------------------------------------------------------------------------
__global__ __launch_bounds__(256) void readout_kernel(
    const float* __restrict__ h, const float* __restrict__ rW,
    float* __restrict__ out)
{
  __shared__ float red[256][3];
  float p0 = 0.0f, p1 = 0.0f, p2 = 0.0f;
  for (int r = threadIdx.x; r < NPT * NORI; r += 256) {
    const float* hr = h + r * HD;
#pragma unroll 8
    for (int c = 0; c < HD; ++c) {
      const float hv = hr[c];
      p0 += hv * rW[c * 3 + 0];
      p1 += hv * rW[c * 3 + 1];
      p2 += hv * rW[c * 3 + 2];
    }
  }
  red[threadIdx.x][0] = p0; red[threadIdx.x][1] = p1; red[threadIdx.x][2] = p2;
  __syncthreads();
  for (int s = 128; s > 0; s >>= 1) {
    if (threadIdx.x < s) {
      red[threadIdx.x][0] += red[threadIdx.x + s][0];
      red[threadIdx.x][1] += red[threadIdx.x + s][1];
      red[threadIdx.x][2] += red[threadIdx.x + s][2];
    }
    __syncthreads();
  }
  if (threadIdx.x < 3) out[threadIdx.x] = red[0][threadIdx.x] * (1.0f / (float)(NPT * NORI));
}

// ---------------------------------------------------------------------------
extern "C" void kernel_launch(void* const* d_in, const int* in_sizes, int n_in,
                              void* d_out, int out_size, void* d_ws, size_t ws_size,
                              hipStream_t stream) {
  const float* pos    = (const float*)d_in[0];
  const float* x      = (const float*)d_in[1];
  const float* skbW1  = (const float*)d_in[2];
  const float* skbB1  = (const float*)d_in[3];
  const float* skbW2  = (const float*)d_in[4];
  const float* skbB2  = (const float*)d_in[5];
  const float* rkbW1  = (const float*)d_in[6];
  const float* rkbB1  = (const float*)d_in[7];
  const float* rkbW2  = (const float*)d_in[8];
  const float* rkbB2  = (const float*)d_in[9];
  const float* embW   = (const float*)d_in[10];
  const float* convSW = (const float*)d_in[11];  // (L,32,64)
  const float* convRW = (const float*)d_in[12];  // (L,32,64)
  const float* convB  = (const float*)d_in[13];  // (L,64)
  const float* lnS    = (const float*)d_in[14];
  const float* lnB    = (const float*)d_in[15];
  const float* lin1W  = (const float*)d_in[16];  // (L,64,256)
  const float* lin1B  = (const float*)d_in[17];  // (L,256)
  const float* lin2W  = (const float*)d_in[18];  // (L,256,64)
  const float* lin2B  = (const float*)d_in[19];  // (L,64)
  const float* rdW    = (const float*)d_in[20];  // (64,3)

  float* ws = (float*)d_ws;
  float* rk   = ws;                         // L*16*16*64 = 32768
  float* hbuf = ws + 32768;                 // N*O*H = 262144
  float* cbuf = ws + 32768 + 262144;        // N*O*H = 262144

  fiber_kernel<<<1, 256, 0, stream>>>(rkbW1, rkbB1, rkbW2, rkbB2, convRW, rk);
  embed_kernel<<<(NPT * NORI * HD) / 256, 256, 0, stream>>>(x, embW, hbuf);

  for (int l = 0; l < NLAY; ++l) {
    conv_kernel<<<(NPT / 4) * NORI, 128, 0, stream>>>(
        pos, skbW1, skbB1, skbW2, skbB2,
        convSW + l * (BDIM * HD), hbuf, cbuf);
    mix_kernel<<<NPT, 32, 0, stream>>>(
        cbuf, rk + l * (NORI * NORI * HD),
        convB + l * HD, lnS + l * HD, lnB + l * HD,
        lin1W + l * (HD * WFH), lin1B + l * WFH,
        lin2W + l * (WFH * HD), lin2B + l * HD, hbuf);
  }

  readout_kernel<<<1, 256, 0, stream>>>(hbuf, rdW, (float*)d_out);
}